// LineRaster2d_24945170055729
// MI455X (gfx1250) — compile-verified
//
#include <hip/hip_runtime.h>

#define H 6144
#define W 6144
#define TILES_PER_DIM (W / 16) /* 384 */
#define WAVES_PER_BLOCK 8
#define TOTAL_WAVES ((H / 16) * (W / 16)) /* 147456 */

typedef float v2f __attribute__((ext_vector_type(2)));
typedef float v8f __attribute__((ext_vector_type(8)));

// One wave32 = one 16x16 tile. V_WMMA_F32_16X16X4_F32 computes the scaled
// projection parameter t_raw(x,y) = (x*d0 + y*d1 - c) / len_sq for all 256
// pixels in one instruction (A row i = [x_i, 1, 0, 0],
// B col j = [d0/L, (y_j*d1 - c)/L, 0, 0]^T). Per-pixel tail is ~5 VALU +
// 1 trans sqrt, stores use literal row offsets folded into the 24-bit
// instruction offset.
__global__ __launch_bounds__(256) void LineRaster2d_wmma_kernel(
    const float* __restrict__ p0in,
    const float* __restrict__ p1in,
    float* __restrict__ out)
{
    // ---- uniform scalar setup ----
    const float p0x = p0in[0] * (float)H;
    const float p0y = p0in[1] * (float)W;
    const float p1x = p1in[0] * (float)H;
    const float p1y = p1in[1] * (float)W;
    const float d0 = p1x - p0x;
    const float d1 = p1y - p0y;
    const float len_sq  = d0 * d0 + d1 * d1;
    const float inv_len = __builtin_amdgcn_rcpf(len_sq);  // 1 trans vs ~10-op div
    const float cterm   = p0x * d0 + p0y * d1;
    const float b_d0    = d0 * inv_len;       // scaled B row 0
    const float b_d1    = d1 * inv_len;       // scaled y coefficient
    const float b_c     = cterm * inv_len;    // scaled constant

    // ---- tile assignment: one wave per 16x16 tile ----
    const unsigned tid  = threadIdx.x;
    const unsigned lane = tid & 31u;
    const unsigned waveId = blockIdx.x * WAVES_PER_BLOCK + (tid >> 5);
    const unsigned tileX  = waveId / TILES_PER_DIM; // row-tile (first axis)
    const unsigned tileY  = waveId % TILES_PER_DIM; // col-tile (contiguous axis)

    const unsigned m16 = lane & 15u;
    const bool lo = lane < 16u;
    const unsigned mOff = lo ? 0u : 8u;
    const float    mOffF = lo ? 0.0f : 8.0f;

    const unsigned rowBaseI = tileX * 16u;
    const unsigned colI     = tileY * 16u + m16;
    const float rowBaseF = (float)rowBaseI;
    const float colF     = (float)colI;

    // A (16x4 f32): lanes 0-15 hold K0/K1 = [x_M, 1]; lanes 16-31 (K2/K3) zero.
    v2f a;
    a.x = lo ? (rowBaseF + (float)m16) : 0.0f;
    a.y = lo ? 1.0f : 0.0f;

    // B (4x16 f32), pre-scaled by 1/len_sq so WMMA outputs t_raw directly.
    v2f b;
    b.x = lo ? b_d0 : 0.0f;
    b.y = lo ? fmaf(colF, b_d1, -b_c) : 0.0f;

    v8f cacc = {};
    // traw[r] = ((x_M*d0 + y_N*d1) - c) / len_sq,  M = r + mOff, N = lane%16
    v8f traw = __builtin_amdgcn_wmma_f32_16x16x4_f32(
        /*neg_a=*/false, a, /*neg_b=*/false, b,
        /*c_mod=*/(short)0, cacc, /*reuse_a=*/false, /*reuse_b=*/false);

    // Per-lane bases (hoisted out of the element loop).
    const float eyBase = colF - p0y;                    // (p - p0)_y
    const float exBase = rowBaseF + mOffF - p0x;        // (p - p0)_x at r = 0
    float* __restrict__ basePtr = out + ((rowBaseI + mOff) * (unsigned)W + colI);

#pragma unroll
    for (int r = 0; r < 8; ++r) {
        float t = fminf(fmaxf(traw[r], 0.0f), 1.0f);    // -> v_max clamp
        const float ex = (exBase + (float)r) - t * d0;  // X - proj_x
        const float ey = eyBase - t * d1;               // Y - proj_y
        const float dist = __builtin_amdgcn_sqrtf(fmaf(ex, ex, ey * ey));
        // dist >= 0, so (dist<1 ? 1-dist : 0) == saturate(1-dist); NaN -> 0 too.
        const float val = fminf(fmaxf(1.0f - dist, 0.0f), 1.0f);
        basePtr[r * W] = val;                           // literal r*W*4 -> ioffset imm
    }
}

extern "C" void kernel_launch(void* const* d_in, const int* in_sizes, int n_in,
                              void* d_out, int out_size, void* d_ws, size_t ws_size,
                              hipStream_t stream) {
    (void)in_sizes; (void)n_in; (void)out_size; (void)d_ws; (void)ws_size;
    const float* p0 = (const float*)d_in[0];
    const float* p1 = (const float*)d_in[1];
    float* out = (float*)d_out;

    const int blocks = TOTAL_WAVES / WAVES_PER_BLOCK; // 18432
    LineRaster2d_wmma_kernel<<<blocks, 256, 0, stream>>>(p0, p1, out);
}